// VectorQuantize_56289841382018
// MI455X (gfx1250) — compile-verified
//
#include <hip/hip_runtime.h>

typedef float v2f __attribute__((ext_vector_type(2)));
typedef float v8f __attribute__((ext_vector_type(8)));

#define N_ROWS   16384
#define DIM      256
#define KCODES   8192
#define NTILES_K 512                 // 8192 / 16
#define IND_OFF  (16384 * 256)       // 4,194,304
#define LOSS_OFF (IND_OFF + 16384)   // 4,210,688
#define PB_FLOATS (KCODES * DIM)     // 2,097,152 floats = 8 MB

// ---------------------------------------------------------------------------
// Prep 1: pack embed [8192,256] into WMMA-B-ready layout.
// For column tile ct (16 codes), K-chunk pair p (chunks c=2p,2p+1 of 4 K each),
// lane l gets a contiguous float4 = {B0(c),B1(c),B0(c+1),B1(c+1)} where
// B0/B1 follow the 16x16x4 f32 B layout: lanes 0-15 hold K=0/1, lanes 16-31
// hold K=2/3, N = lane&15.
// ---------------------------------------------------------------------------
__global__ void vq_pack_b(const float* __restrict__ embed, float* __restrict__ pb) {
    unsigned t  = blockIdx.x * blockDim.x + threadIdx.x;   // 0 .. 2M-1
    unsigned q  = t & 3u;
    unsigned l  = (t >> 2) & 31u;
    unsigned p  = (t >> 7) & 31u;
    unsigned ct = t >> 12;
    unsigned c   = 2u * p + (q >> 1);
    unsigned col = ct * 16u + (l & 15u);
    unsigned k   = 4u * c + ((l & 16u) ? 2u : 0u) + (q & 1u);
    pb[t] = embed[col * 256u + k];
}

// ---------------------------------------------------------------------------
// Prep 2: ||e_j||^2 per codebook row (one wave per row), and zero loss slot.
// ---------------------------------------------------------------------------
__global__ void vq_norms(const float* __restrict__ embed, float* __restrict__ enorm,
                         float* __restrict__ loss_slot) {
    if (blockIdx.x == 0 && threadIdx.x == 0) *loss_slot = 0.0f;
    int lane = threadIdx.x & 31;
    int w    = threadIdx.x >> 5;
    int row  = blockIdx.x * 8 + w;                 // 1024 blocks * 8 waves = 8192 rows
    const float4* r4 = (const float4*)(embed + (size_t)row * 256);
    float4 a = r4[2 * lane];
    float4 b = r4[2 * lane + 1];
    float s = a.x*a.x + a.y*a.y + a.z*a.z + a.w*a.w
            + b.x*b.x + b.y*b.y + b.z*b.z + b.w*b.w;
    #pragma unroll
    for (int m = 1; m < 32; m <<= 1) s += __shfl_xor(s, m, 32);
    if (lane == 0) enorm[row] = s;
}

// ---------------------------------------------------------------------------
// Main: one wave per 32-row tile of x (two 16-row WMMA A fragments held in
// 256 VGPRs). Each packed-B float4 load feeds FOUR f32 WMMAs (two row groups
// x two K-chunks), so B-side L2 traffic is 4 GB total and the loop is
// matrix-issue bound. Running argmax per lane, butterfly reduce, gather +
// commit loss.
// ---------------------------------------------------------------------------
__global__ __launch_bounds__(128) void vq_main(const float* __restrict__ x,
                                               const float* __restrict__ embed,
                                               const float* __restrict__ pb,
                                               const float* __restrict__ enorm,
                                               float* __restrict__ out) {
    const int lane = threadIdx.x & 31;
    const int wave = blockIdx.x * (blockDim.x >> 5) + (threadIdx.x >> 5);
    const int m0   = wave * 32;                    // 512 waves cover 16384 rows

    float* quant = out;
    float* ind   = out + IND_OFF;
    float* loss  = out + LOSS_OFF;

    // ---- load two A tiles (rows m0..m0+15 and m0+16..m0+31), WMMA A layout
    // lanes 0-15: K = 4c+0,4c+1 ; lanes 16-31: K = 4c+2,4c+3 ; M = lane&15
    v2f A0[64], A1[64];
    {
        const float* xr0 = x + (size_t)(m0 + (lane & 15)) * 256 + ((lane >> 4) << 1);
        const float* xr1 = xr0 + 16 * 256;
        #pragma unroll
        for (int c = 0; c < 64; ++c) {
            A0[c] = *(const v2f*)(xr0 + 4 * c);
            A1[c] = *(const v2f*)(xr1 + 4 * c);
        }
    }

    float bestv0[8], bestv1[8];
    int   besti0[8], besti1[8];
    #pragma unroll
    for (int r = 0; r < 8; ++r) {
        bestv0[r] = -3.4e38f; besti0[r] = 0;
        bestv1[r] = -3.4e38f; besti1[r] = 0;
    }

    #pragma unroll 1
    for (int ct = 0; ct < NTILES_K; ++ct) {
        const float4* pbt = (const float4*)(pb + (size_t)ct * 4096) + lane;
        __builtin_prefetch(pb + (size_t)(ct + 1) * 4096 + lane * 4, 0, 1);

        v8f c00 = {}, c01 = {};    // row group 0, even/odd K-chunk accumulators
        v8f c10 = {}, c11 = {};    // row group 1
        #pragma unroll
        for (int p = 0; p < 32; ++p) {
            float4 b4 = pbt[p * 32];
            v2f b0; b0.x = b4.x; b0.y = b4.y;
            v2f b1; b1.x = b4.z; b1.y = b4.w;
            c00 = __builtin_amdgcn_wmma_f32_16x16x4_f32(
                      false, A0[2 * p],     false, b0, (short)0, c00, false, false);
            c10 = __builtin_amdgcn_wmma_f32_16x16x4_f32(
                      false, A1[2 * p],     false, b0, (short)0, c10, false, false);
            c01 = __builtin_amdgcn_wmma_f32_16x16x4_f32(
                      false, A0[2 * p + 1], false, b1, (short)0, c01, false, false);
            c11 = __builtin_amdgcn_wmma_f32_16x16x4_f32(
                      false, A1[2 * p + 1], false, b1, (short)0, c11, false, false);
        }

        int   colidx = ct * 16 + (lane & 15);
        float en     = enorm[colidx];
        #pragma unroll
        for (int r = 0; r < 8; ++r) {
            float s0 = 2.0f * (c00[r] + c01[r]) - en;   // ||x||^2 dropped (row const)
            float s1 = 2.0f * (c10[r] + c11[r]) - en;
            if (s0 > bestv0[r]) { bestv0[r] = s0; besti0[r] = colidx; }
            if (s1 > bestv1[r]) { bestv1[r] = s1; besti1[r] = colidx; }
        }
    }

    // ---- argmax reduction across the 16 lanes of each half-wave ----------
    #pragma unroll
    for (int m = 1; m <= 8; m <<= 1) {
        #pragma unroll
        for (int r = 0; r < 8; ++r) {
            float ov0 = __shfl_xor(bestv0[r], m, 32);
            int   oi0 = __shfl_xor(besti0[r], m, 32);
            if (ov0 > bestv0[r] || (ov0 == bestv0[r] && oi0 < besti0[r])) {
                bestv0[r] = ov0; besti0[r] = oi0;
            }
            float ov1 = __shfl_xor(bestv1[r], m, 32);
            int   oi1 = __shfl_xor(besti1[r], m, 32);
            if (ov1 > bestv1[r] || (ov1 == bestv1[r] && oi1 < besti1[r])) {
                bestv1[r] = ov1; besti1[r] = oi1;
            }
        }
    }

    // ---- write indices (as float, matching flattened f32 output) ---------
    if ((lane & 15) == 0) {
        int h = lane >> 4;
        #pragma unroll
        for (int r = 0; r < 8; ++r) {
            ind[m0 +      r + 8 * h] = (float)besti0[r];
            ind[m0 + 16 + r + 8 * h] = (float)besti1[r];
        }
    }

    // ---- gather quantized rows + commit loss ------------------------------
    float acc = 0.0f;
    for (int q = 0; q < 32; ++q) {
        int qq   = q & 15;                          // row within group
        int src  = (qq >> 3) << 4;                  // lane holding that row's best
        int idxq = (q < 16) ? __shfl(besti0[qq & 7], src, 32)
                            : __shfl(besti1[qq & 7], src, 32);
        const float4* e4 = (const float4*)(embed + (size_t)idxq * 256);
        const float4* x4 = (const float4*)(x + (size_t)(m0 + q) * 256);
        float4*       o4 = (float4*)(quant + (size_t)(m0 + q) * 256);
        #pragma unroll
        for (int v = 0; v < 2; ++v) {
            float4 e  = e4[2 * lane + v];
            float4 xv = x4[2 * lane + v];
            o4[2 * lane + v] = e;
            float dx = e.x - xv.x, dy = e.y - xv.y;
            float dz = e.z - xv.z, dw = e.w - xv.w;
            acc += dx * dx + dy * dy + dz * dz + dw * dw;
        }
    }
    #pragma unroll
    for (int m = 1; m < 32; m <<= 1) acc += __shfl_xor(acc, m, 32);
    if (lane == 0) atomicAdd(loss, acc * (1.0f / 4194304.0f));
}

// ---------------------------------------------------------------------------
extern "C" void kernel_launch(void* const* d_in, const int* in_sizes, int n_in,
                              void* d_out, int out_size, void* d_ws, size_t ws_size,
                              hipStream_t stream) {
    (void)in_sizes; (void)n_in; (void)out_size; (void)ws_size;
    const float* x     = (const float*)d_in[0];
    const float* embed = (const float*)d_in[1];
    float* pb    = (float*)d_ws;                 // 8 MB packed B
    float* enorm = pb + PB_FLOATS;               // + 32 KB norms
    float* out   = (float*)d_out;

    vq_pack_b<<<PB_FLOATS / 256, 256, 0, stream>>>(embed, pb);
    vq_norms <<<1024, 256, 0, stream>>>(embed, enorm, out + LOSS_OFF);
    vq_main  <<<128, 128, 0, stream>>>(x, embed, pb, enorm, out);
}